// RGCNModel_56169582297302
// MI455X (gfx1250) — compile-verified
//
#include <hip/hip_runtime.h>
#include <hip/hip_bf16.h>

typedef __attribute__((ext_vector_type(2))) float v2f;
typedef __attribute__((ext_vector_type(8))) float v8f;

#define D 128
#define TM 64           // rows per block in GEMM
#define LDSTRIDE 132    // padded LDS row stride (floats)

// ---------------------------------------------------------------------------
// Build relation weights directly in WMMA-B fragment-packed order.
// Packed index for (kstep = k/4, wave = col/16, lane, slot):
//   p = (kstep/2)*1024 + wave*128 + lane*4 + (kstep&1)*2 + slot
// holding B[4*kstep + 2*(lane>>4) + slot][wave*16 + (lane&15)], so one
// float4 load per lane supplies the B fragments of two consecutive k-steps.
// ---------------------------------------------------------------------------
__global__ __launch_bounds__(256) void compute_w_kernel(
    const float* __restrict__ comp, const float* __restrict__ bases,
    const float* __restrict__ root, float* __restrict__ Wpk,
    float* __restrict__ W8pk, int layer)
{
    int idx = blockIdx.x * 256 + threadIdx.x;      // < 9 * 16384
    int r = idx >> 14;
    int p = idx & 16383;
    // unpack packed index -> (k, col)
    int pair  = p >> 10;
    int rem   = p & 1023;
    int wave  = rem >> 7;
    int rem2  = rem & 127;
    int lane  = rem2 >> 2;
    int sl2   = rem2 & 3;
    int kstep = pair * 2 + (sl2 >> 1);
    int slot  = sl2 & 1;
    int k   = kstep * 4 + (lane >> 4) * 2 + slot;
    int col = wave * 16 + (lane & 15);
    int ij  = k * D + col;

    const float* cp = comp + ((size_t)layer * 9 + r) * 4;
    const float* bp = bases + (size_t)layer * 4 * 16384;
    float s = cp[0] * bp[ij]
            + cp[1] * bp[16384 + ij]
            + cp[2] * bp[32768 + ij]
            + cp[3] * bp[49152 + ij];
    Wpk[idx] = s;
    if (r == 8) W8pk[p] = s + root[(size_t)layer * 16384 + ij];
}

// ---------------------------------------------------------------------------
// relation histogram (block-local LDS histogram -> 8 global atomics/block)
// ---------------------------------------------------------------------------
__global__ __launch_bounds__(256) void hist_kernel(
    const long long* __restrict__ et, int* __restrict__ relcnt, long long E)
{
    __shared__ int lh[8];
    if (threadIdx.x < 8) lh[threadIdx.x] = 0;
    __syncthreads();
    long long e = (long long)blockIdx.x * 256 + threadIdx.x;
    if (e < E) atomicAdd(&lh[(int)et[e]], 1);
    __syncthreads();
    if (threadIdx.x < 8) atomicAdd(&relcnt[threadIdx.x], lh[threadIdx.x]);
}

__global__ void prefix_kernel(const int* __restrict__ relcnt,
                              int* __restrict__ offs, int* __restrict__ cursor)
{
    if (blockIdx.x == 0 && threadIdx.x == 0) {
        int s = 0;
        for (int r = 0; r < 8; ++r) { offs[r] = s; cursor[r] = s; s += relcnt[r]; }
        offs[8] = s;
    }
}

// counting-sort edges by relation into int32 src/dst arrays
__global__ __launch_bounds__(256) void reorder_kernel(
    const long long* __restrict__ ei, const long long* __restrict__ et,
    int* __restrict__ cursor, int* __restrict__ srcs, int* __restrict__ dsts,
    long long E)
{
    __shared__ int lh[8];
    __shared__ int lbase[8];
    if (threadIdx.x < 8) lh[threadIdx.x] = 0;
    __syncthreads();
    long long e = (long long)blockIdx.x * 256 + threadIdx.x;
    int r = 0, lpos = 0;
    bool valid = (e < E);
    if (valid) { r = (int)et[e]; lpos = atomicAdd(&lh[r], 1); }
    __syncthreads();
    if (threadIdx.x < 8) lbase[threadIdx.x] = atomicAdd(&cursor[threadIdx.x], lh[threadIdx.x]);
    __syncthreads();
    if (valid) {
        int pos = lbase[r] + lpos;
        srcs[pos] = (int)ei[e];
        dsts[pos] = (int)ei[E + e];
    }
}

// ---------------------------------------------------------------------------
// per-(relation,dst) edge counts (real edges only; self-loop cnt == 1)
// ---------------------------------------------------------------------------
__global__ __launch_bounds__(256) void count_kernel(
    const long long* __restrict__ ei, const long long* __restrict__ et,
    float* __restrict__ cnt, long long E, int N)
{
    long long e = (long long)blockIdx.x * 256 + threadIdx.x;
    if (e >= E) return;
    int r   = (int)et[e];
    int dst = (int)ei[E + e];
    atomicAdd(&cnt[(size_t)r * N + dst], 1.0f);
}

__global__ __launch_bounds__(256) void invert_kernel(float* __restrict__ cnt, long long n)
{
    long long i = (long long)blockIdx.x * 256 + threadIdx.x;
    if (i >= n) return;
    float c = cnt[i];
    cnt[i] = (c > 0.0f) ? (1.0f / c) : 0.0f;
}

// ---------------------------------------------------------------------------
// persistent scatter over this relation's sorted edge range:
// one wave per edge iteration, pre[dst,:] += h[src,:]
// ---------------------------------------------------------------------------
__global__ __launch_bounds__(256) void scatter_sorted_kernel(
    const int* __restrict__ srcs, const int* __restrict__ dsts,
    const int* __restrict__ offs, const float* __restrict__ h,
    float* __restrict__ pre, int rel)
{
    const int lane = threadIdx.x & 31;
    long long wid = (((long long)blockIdx.x * 256) + threadIdx.x) >> 5;
    long long nw  = ((long long)gridDim.x * 256) >> 5;
    const int lo = offs[rel];
    const int hiE = offs[rel + 1];
    for (long long e = lo + wid; e < hiE; e += nw) {
        int src = srcs[e];
        int dst = dsts[e];
        const float4 v = *(const float4*)(h + (size_t)src * D + lane * 4);
        float* p = pre + (size_t)dst * D + lane * 4;
        atomicAdd(p + 0, v.x);
        atomicAdd(p + 1, v.y);
        atomicAdd(p + 2, v.z);
        atomicAdd(p + 3, v.w);
    }
}

// ---------------------------------------------------------------------------
// C[row,:] (+)= (A[row,:] * rowscale[row]) @ B   (A:[N,128], Bpk packed 128x128)
// initFromBias : acc init from bias[col] (or 0); else from C.
// epimode: 0 = plain store; 1 = store 0.5*hprev + 0.5*relu(acc);
//          2 = store 0.5*hprev + 0.5*acc.
// Block: 256 threads = 8 waves; 64 rows x 128 cols; wave w -> cols 16w..16w+15,
// 4 row-tiles of 16 rows. fp32 WMMA 16x16x4.
// ---------------------------------------------------------------------------
__global__ __launch_bounds__(256) void gemm_wmma_kernel(
    const float* __restrict__ A, const float* __restrict__ rowscale,
    const float* __restrict__ Bpk, float* __restrict__ C,
    const float* __restrict__ bias, const float* __restrict__ hprev,
    int initFromBias, int epimode, int N)
{
    __shared__ float sA[TM * LDSTRIDE];
    const int tid  = threadIdx.x;
    const int lane = tid & 31;
    const int wave = tid >> 5;
    const int hi   = lane >> 4;
    const int l16  = lane & 15;
    const int row0 = blockIdx.x * TM;
    const int col0 = wave * 16;

    // cooperative, coalesced A-tile load with optional per-row scaling
    #pragma unroll
    for (int t = 0; t < 8; ++t) {
        int r  = t * 8 + (tid >> 5);
        int c4 = (tid & 31) * 4;
        int gr = row0 + r;
        if (gr > N - 1) gr = N - 1;    // clamp; masked at store
        float4 v = *(const float4*)(A + (size_t)gr * D + c4);
        float s = rowscale ? rowscale[gr] : 1.0f;
        v.x *= s; v.y *= s; v.z *= s; v.w *= s;
        *(float4*)(&sA[r * LDSTRIDE + c4]) = v;
    }
    __syncthreads();

    v8f acc[4];
    #pragma unroll
    for (int rt = 0; rt < 4; ++rt) {
        v8f c;
        #pragma unroll
        for (int i = 0; i < 8; ++i) {
            int row = row0 + rt * 16 + i + 8 * hi;   // C layout: VGPR i -> M=i / i+8
            int col = col0 + l16;
            float val;
            if (initFromBias) val = bias ? bias[col] : 0.0f;
            else              val = (row < N) ? C[(size_t)row * D + col] : 0.0f;
            c[i] = val;
        }
        acc[rt] = c;
    }

    const float4* Bp4 = (const float4*)Bpk;
    const int bidx = wave * 32 + lane;
    for (int k = 0; k < D; k += 8) {
        float4 bb = Bp4[(k >> 3) * 256 + bidx];      // B frags for 2 k-steps
        v2f b0; b0.x = bb.x; b0.y = bb.y;
        v2f b1; b1.x = bb.z; b1.y = bb.w;
        int kb0 = k + 2 * hi;
        #pragma unroll
        for (int rt = 0; rt < 4; ++rt) {
            const float* ap = &sA[(rt * 16 + l16) * LDSTRIDE + kb0];
            v2f a; a.x = ap[0]; a.y = ap[1];
            acc[rt] = __builtin_amdgcn_wmma_f32_16x16x4_f32(
                false, a, false, b0, (short)0, acc[rt], false, false);
        }
        int kb1 = kb0 + 4;
        #pragma unroll
        for (int rt = 0; rt < 4; ++rt) {
            const float* ap = &sA[(rt * 16 + l16) * LDSTRIDE + kb1];
            v2f a; a.x = ap[0]; a.y = ap[1];
            acc[rt] = __builtin_amdgcn_wmma_f32_16x16x4_f32(
                false, a, false, b1, (short)0, acc[rt], false, false);
        }
    }

    #pragma unroll
    for (int rt = 0; rt < 4; ++rt) {
        #pragma unroll
        for (int i = 0; i < 8; ++i) {
            int row = row0 + rt * 16 + i + 8 * hi;
            int col = col0 + l16;
            if (row < N) {
                float val = acc[rt][i];
                if (epimode) {
                    if (epimode == 1) val = fmaxf(val, 0.0f);
                    val = 0.5f * hprev[(size_t)row * D + col] + 0.5f * val;
                }
                C[(size_t)row * D + col] = val;
            }
        }
    }
}

// ---------------------------------------------------------------------------
extern "C" void kernel_launch(void* const* d_in, const int* in_sizes, int n_in,
                              void* d_out, int out_size, void* d_ws, size_t ws_size,
                              hipStream_t stream)
{
    const float*     x     = (const float*)d_in[0];
    const long long* ei    = (const long long*)d_in[1];   // [2,E] int64
    const long long* et    = (const long long*)d_in[2];   // [E]   int64
    const float*     bases = (const float*)d_in[3];       // [HOP,4,128,128]
    const float*     comp  = (const float*)d_in[4];       // [HOP,9,4]
    const float*     root  = (const float*)d_in[5];       // [HOP,128,128]
    const float*     bias  = (const float*)d_in[6];       // [HOP,128]

    const int       N   = in_sizes[0] / D;
    const long long E   = (long long)in_sizes[1] / 2;
    const int       HOP = in_sizes[3] / (4 * D * D);

    // workspace layout (256-byte aligned slices)
    char*  base = (char*)d_ws;
    size_t off  = 0;
    auto alloc = [&](size_t bytes) {
        void* p = base + off;
        off += (bytes + 255) & ~(size_t)255;
        return p;
    };
    float* pre    = (float*)alloc((size_t)N * D * 4);   // per-relation agg buffer
    float* cnt    = (float*)alloc((size_t)8 * N * 4);   // inverse (rel,dst) counts
    float* Wpk    = (float*)alloc((size_t)9 * D * D * 4);
    float* W8pk   = (float*)alloc((size_t)D * D * 4);
    int*   srcs   = (int*)alloc((size_t)E * 4);
    int*   dsts   = (int*)alloc((size_t)E * 4);
    int*   relcnt = (int*)alloc(64);
    int*   offs   = (int*)alloc(64);
    int*   cursor = (int*)alloc(64);
    float* hA     = (float*)alloc((size_t)N * D * 4);   // hidden ping
    float* hB     = (float*)alloc((size_t)N * D * 4);   // pong (HOP>2 only)

    const int      gemm_grid = (N + TM - 1) / TM;
    const unsigned edge_grid = (unsigned)((E + 255) / 256);
    const unsigned scat_grid = 4096;                    // persistent scatter waves

    // --- one-time edge preprocessing (shared across layers) ---
    hipMemsetAsync(relcnt, 0, 32, stream);
    hist_kernel<<<edge_grid, 256, 0, stream>>>(et, relcnt, E);
    prefix_kernel<<<1, 32, 0, stream>>>(relcnt, offs, cursor);
    reorder_kernel<<<edge_grid, 256, 0, stream>>>(ei, et, cursor, srcs, dsts, E);

    hipMemsetAsync(cnt, 0, (size_t)8 * N * 4, stream);
    count_kernel<<<edge_grid, 256, 0, stream>>>(ei, et, cnt, E, N);
    invert_kernel<<<(unsigned)(((long long)8 * N + 255) / 256), 256, 0, stream>>>(
        cnt, (long long)8 * N);

    for (int l = 0; l < HOP; ++l) {
        const float* hcur = (l == 0) ? x : ((l & 1) ? hA : hB);
        float*       agg  = (l == HOP - 1) ? (float*)d_out : ((l & 1) ? hB : hA);
        const int    epimode = (l < HOP - 1) ? 1 : 2;    // relu+blend / blend

        compute_w_kernel<<<(9 * D * D) / 256, 256, 0, stream>>>(
            comp, bases, root, Wpk, W8pk, l);

        // agg = h @ (W[8] + root) + bias   (self-loop cnt == 1)
        gemm_wmma_kernel<<<gemm_grid, 256, 0, stream>>>(
            hcur, nullptr, W8pk, agg, bias + (size_t)l * D, nullptr, 1, 0, N);

        for (int r = 0; r < 8; ++r) {
            hipMemsetAsync(pre, 0, (size_t)N * D * 4, stream);
            scatter_sorted_kernel<<<scat_grid, 256, 0, stream>>>(
                srcs, dsts, offs, hcur, pre, r);
            // agg += (pre * invcnt_r) @ W[r]; last relation fuses the epilogue
            gemm_wmma_kernel<<<gemm_grid, 256, 0, stream>>>(
                pre, cnt + (size_t)r * N, Wpk + (size_t)r * D * D, agg,
                nullptr, hcur, 0, (r == 7) ? epimode : 0, N);
        }
    }
}